// TransitionDown_39728447488678
// MI455X (gfx1250) — compile-verified
//
#include <hip/hip_runtime.h>
#include <hip/hip_bf16.h>

// ---------------------------------------------------------------------------
// TransitionDown (vector-neuron) for MI455X / gfx1250, wave32.
//   Stage 1: FPS      - single workgroup, register-resident points, 2 barriers/iter
//   Stage 2: KNN      - 1 thread/query, unrolled 16-entry register insertion sort
//   Stage 3: VN GEMM  - V_WMMA_F32_16X16X4_F32 (full f32 precision),
//                       wave = one 16-out-channel tile of one query point,
//                       3 vector components = 3 B-matrices -> lane-local LeakyReLU
// ---------------------------------------------------------------------------

#define NPTS      16384
#define MOUT      4096      // NPTS / STRIDE
#define KNN_K     16
#define CIN       64
#define COUT      64
#define NEG_SLOPE 0.2f
#define VN_EPS    1e-6f

typedef __attribute__((ext_vector_type(2))) float v2f;
typedef __attribute__((ext_vector_type(8))) float v8f;

static __device__ __forceinline__ v8f wmma_f32_k4(v2f a, v2f b, v8f c) {
    // D(16x16,f32) += A(16x4,f32) * B(4x16,f32)
    return __builtin_amdgcn_wmma_f32_16x16x4_f32(
        /*neg_a=*/false, a, /*neg_b=*/false, b,
        /*c_mod=*/(short)0, c, /*reuse_a=*/false, /*reuse_b=*/false);
}

// ---------------------------------------------------------------------------
// Stage 1: furthest point sampling. One block of 1024 threads (32 waves).
// Each thread owns 16 points (strided for coalescing); d2 lives in registers.
// ---------------------------------------------------------------------------
__global__ __launch_bounds__(1024) void fps_kernel(const float* __restrict__ p,
                                                   int* __restrict__ fps_idx) {
    const int t = threadIdx.x;
    const int T = 1024;
    const int J = NPTS / T;   // 16 points per thread

    float px[J], py[J], pz[J], dd[J];
    const float q0x = p[0], q0y = p[1], q0z = p[2];
#pragma unroll
    for (int j = 0; j < J; ++j) {
        const int n = t + j * T;           // strided -> coalesced loads
        px[j] = p[3 * n + 0];
        py[j] = p[3 * n + 1];
        pz[j] = p[3 * n + 2];
        const float dx = px[j] - q0x, dy = py[j] - q0y, dz = pz[j] - q0z;
        dd[j] = dx * dx + dy * dy + dz * dz;
    }
    if (t == 0) fps_idx[0] = 0;

    __shared__ float sval[32];
    __shared__ int   sidx[32];
    __shared__ int   swin;

    const int lane = t & 31;
    const int wv   = t >> 5;

    for (int i = 1; i < MOUT; ++i) {
        // thread-local argmax (first-max wins -> smallest index on ties)
        float best = -1.0f; int bidx = 0;
#pragma unroll
        for (int j = 0; j < J; ++j) {
            if (dd[j] > best) { best = dd[j]; bidx = t + j * T; }
        }
        // wave32 shuffle argmax reduction
#pragma unroll
        for (int mask = 16; mask >= 1; mask >>= 1) {
            const float ov = __shfl_xor(best, mask, 32);
            const int   oi = __shfl_xor(bidx, mask, 32);
            if (ov > best || (ov == best && oi < bidx)) { best = ov; bidx = oi; }
        }
        if (lane == 0) { sval[wv] = best; sidx[wv] = bidx; }
        __syncthreads();
        if (wv == 0) {
            float bv = sval[lane]; int bi = sidx[lane];
#pragma unroll
            for (int mask = 16; mask >= 1; mask >>= 1) {
                const float ov = __shfl_xor(bv, mask, 32);
                const int   oi = __shfl_xor(bi, mask, 32);
                if (ov > bv || (ov == bv && oi < bi)) { bv = ov; bi = oi; }
            }
            if (lane == 0) { swin = bi; fps_idx[i] = bi; }
        }
        __syncthreads();
        const int nxt = swin;              // broadcast: all lanes read same addr
        const float qx = p[3 * nxt + 0], qy = p[3 * nxt + 1], qz = p[3 * nxt + 2];
#pragma unroll
        for (int j = 0; j < J; ++j) {
            const float dx = px[j] - qx, dy = py[j] - qy, dz = pz[j] - qz;
            const float d = dx * dx + dy * dy + dz * dz;
            dd[j] = fminf(dd[j], d);
        }
        // next sval write only happens after every thread passed the barrier
        // above and consumed swin -> 2 barriers per iteration suffice.
    }
}

// ---------------------------------------------------------------------------
// Stage 2: brute-force 16-NN. One thread per query; sorted insertion kept in
// registers via fully unrolled compile-time-indexed swap chain.
// Also emits n_p (downsampled coords) and n_o.
// ---------------------------------------------------------------------------
__global__ __launch_bounds__(256) void knn_kernel(const float* __restrict__ p,
                                                  const int* __restrict__ fps_idx,
                                                  int* __restrict__ nn,
                                                  float* __restrict__ n_p,
                                                  int* __restrict__ n_o) {
    const int m = blockIdx.x * blockDim.x + threadIdx.x;
    if (m >= MOUT) return;
    const int qi = fps_idx[m];
    const float qx = p[3 * qi + 0], qy = p[3 * qi + 1], qz = p[3 * qi + 2];
    n_p[3 * m + 0] = qx;
    n_p[3 * m + 1] = qy;
    n_p[3 * m + 2] = qz;
    if (m == 0) *n_o = MOUT;

    float bd[KNN_K]; int bi[KNN_K];
#pragma unroll
    for (int s = 0; s < KNN_K; ++s) { bd[s] = 3.4e38f; bi[s] = 0; }

    for (int n = 0; n < NPTS; ++n) {
        const float dx = p[3 * n + 0] - qx;
        const float dy = p[3 * n + 1] - qy;
        const float dz = p[3 * n + 2] - qz;
        float v = dx * dx + dy * dy + dz * dz;
        if (v < bd[KNN_K - 1]) {
            int id = n;
#pragma unroll
            for (int s = 0; s < KNN_K; ++s) {   // bubble-down insertion
                if (v < bd[s]) {
                    const float tv = bd[s]; const int ti = bi[s];
                    bd[s] = v; bi[s] = id; v = tv; id = ti;
                }
            }
        }
    }
#pragma unroll
    for (int s = 0; s < KNN_K; ++s) nn[m * KNN_K + s] = bi[s];
}

// ---------------------------------------------------------------------------
// Stage 3: VN linear + equivariant LeakyReLU + mean over k, via f32 WMMA.
// Block = 256 threads = 8 waves = 2 query points x 4 Cout tiles.
// Per wave: D[16x16] tiles (rows = Cout slice, cols = 16 neighbors),
//           one accumulator set per vector component i in {0,1,2},
//           for q (W_feat) and d (W_dir): 6 x v8f accumulators.
// K = Cin = 64 -> 16 x k4 WMMA steps per accumulator (96 WMMAs / wave).
// ---------------------------------------------------------------------------
__global__ __launch_bounds__(256) void vn_wmma_kernel(const float* __restrict__ x,
                                                      const float* __restrict__ Wf,
                                                      const float* __restrict__ Wd,
                                                      const int* __restrict__ nn,
                                                      float* __restrict__ xout) {
    const int tid  = threadIdx.x;
    const int wave = tid >> 5;
    const int lane = tid & 31;
    const int r    = lane & 15;      // col (neighbor) / A row
    const int h    = lane >> 4;      // half: selects K pair inside WMMA operands

    const int m  = blockIdx.x * 2 + (wave >> 2);   // query point
    const int ot = wave & 3;                       // Cout tile (16 rows)
    const int orow = ot * 16 + r;

    // Hoist A operands (weights) out of the K loop: ISA 16x4 f32 A layout,
    // lane(r,h): VGPR0 = A[r, 2h], VGPR1 = A[r, 2h+1].
    v2f af[16], ad[16];
#pragma unroll
    for (int kk = 0; kk < 16; ++kk) {
        const int c0 = kk * 4 + 2 * h;
        v2f a;
        a.x = Wf[orow * CIN + c0];
        a.y = Wf[orow * CIN + c0 + 1];
        af[kk] = a;
        a.x = Wd[orow * CIN + c0];
        a.y = Wd[orow * CIN + c0 + 1];
        ad[kk] = a;
    }

    // This wave's column r maps to neighbor nn[m*16 + r].
    const int nb = nn[m * KNN_K + r];
    const float* xb = x + (size_t)nb * (CIN * 3);

    v8f q0 = {}, q1 = {}, q2 = {}, d0 = {}, d1 = {}, d2 = {};
#pragma unroll
    for (int kk = 0; kk < 16; ++kk) {
        const int c0 = kk * 4 + 2 * h;          // two channels handled per half
        // 6 consecutive floats: channels c0,c0+1 each with 3 vector comps.
        const float g0x = xb[c0 * 3 + 0], g0y = xb[c0 * 3 + 1], g0z = xb[c0 * 3 + 2];
        const float g1x = xb[c0 * 3 + 3], g1y = xb[c0 * 3 + 4], g1z = xb[c0 * 3 + 5];
        // B 4x16 f32 layout: lane(r,h): VGPR0 = B[2h, r], VGPR1 = B[2h+1, r].
        v2f b0, b1, b2;
        b0.x = g0x; b0.y = g1x;
        b1.x = g0y; b1.y = g1y;
        b2.x = g0z; b2.y = g1z;
        q0 = wmma_f32_k4(af[kk], b0, q0);
        q1 = wmma_f32_k4(af[kk], b1, q1);
        q2 = wmma_f32_k4(af[kk], b2, q2);
        d0 = wmma_f32_k4(ad[kk], b0, d0);
        d1 = wmma_f32_k4(ad[kk], b1, d1);
        d2 = wmma_f32_k4(ad[kk], b2, d2);
    }

    // Equivariant LeakyReLU, lane-local because the 3 components share lane
    // positions across accumulator sets:
    //   out_i = q_i               (dot >= 0)
    //   out_i = q_i - 0.8*(dot/dns)*d_i   (dot < 0), 0.8 = 1 - NEG_SLOPE
    v8f o0, o1, o2;
#pragma unroll
    for (int e = 0; e < 8; ++e) {
        const float qa = q0[e], qb = q1[e], qc = q2[e];
        const float da = d0[e], db = d1[e], dc = d2[e];
        const float dot = qa * da + qb * db + qc * dc;
        const float dns = da * da + db * db + dc * dc + VN_EPS;
        const float t   = (dot >= 0.0f) ? 0.0f : (dot / dns);
        const float u   = (1.0f - NEG_SLOPE) * t;
        o0[e] = qa - u * da;
        o1[e] = qb - u * db;
        o2[e] = qc - u * dc;
    }

    // Mean over the 16 neighbor columns: C/D layout row = e + 8h, col = lane%16.
    // Reduce within each 16-lane half (wave32 shfl_xor, masks 1..8).
#pragma unroll
    for (int e = 0; e < 8; ++e) {
        float a = o0[e], b = o1[e], c = o2[e];
#pragma unroll
        for (int mask = 1; mask <= 8; mask <<= 1) {
            a += __shfl_xor(a, mask, 32);
            b += __shfl_xor(b, mask, 32);
            c += __shfl_xor(c, mask, 32);
        }
        if (r == e) {   // lane e (low half) writes row e; lane 16+e writes row e+8
            const int o = ot * 16 + e + 8 * h;
            // x_out layout (1, Cout, 3, M) -> ((o*3)+i)*M + m
            xout[((size_t)(o * 3 + 0)) * MOUT + m] = a * (1.0f / KNN_K);
            xout[((size_t)(o * 3 + 1)) * MOUT + m] = b * (1.0f / KNN_K);
            xout[((size_t)(o * 3 + 2)) * MOUT + m] = c * (1.0f / KNN_K);
        }
    }
}

// ---------------------------------------------------------------------------
extern "C" void kernel_launch(void* const* d_in, const int* in_sizes, int n_in,
                              void* d_out, int out_size, void* d_ws, size_t ws_size,
                              hipStream_t stream) {
    const float* p  = (const float*)d_in[0];   // (N,3)
    const float* x  = (const float*)d_in[1];   // (N,64,3)
    const float* Wf = (const float*)d_in[2];   // (64,64)
    const float* Wd = (const float*)d_in[3];   // (64,64)
    (void)in_sizes; (void)n_in; (void)out_size; (void)ws_size;

    int*   fps_idx = (int*)d_ws;               // MOUT ints
    int*   nn      = fps_idx + MOUT;           // MOUT*16 ints

    float* n_p  = (float*)d_out;                                   // (M,3)
    float* xout = n_p + (size_t)MOUT * 3;                          // (1,64,3,M)
    int*   n_o  = (int*)(xout + (size_t)COUT * 3 * MOUT);          // (1,) int32

    fps_kernel<<<1, 1024, 0, stream>>>(p, fps_idx);
    knn_kernel<<<(MOUT + 255) / 256, 256, 0, stream>>>(p, fps_idx, nn, n_p, n_o);
    vn_wmma_kernel<<<MOUT / 2, 256, 0, stream>>>(x, Wf, Wd, nn, xout);
}